// BeliefStep_7765300871585
// MI455X (gfx1250) — compile-verified
//
#include <hip/hip_runtime.h>
#include <math.h>

typedef float v4f __attribute__((ext_vector_type(4)));

#define BLK 256
#define DT_         0.1f
#define PI_         3.14159265358979323846f
#define TWO_PI_     6.28318530717958647692f
#define INV_TWO_PI_ 0.15915494309189533577f
#define EXP_M8_     0.00033546262790251185f  // exp(-8)

// --- CDNA5 async global->LDS copy (ISA 15.18.3 op 98, tracked by ASYNCcnt).
// Per-lane: copies 16 bytes from global address vaddr to LDS byte offset vdst.
// The low 32 bits of a generic pointer to __shared__ are the workgroup-
// relative LDS byte offset (AS3 value), which is what VDST must hold.
__device__ __forceinline__ void async_load_b128_to_lds(const void* gptr,
                                                       void* lds_slot) {
  unsigned lds_addr = (unsigned)(unsigned long long)(uintptr_t)lds_slot;
  unsigned long long ga = (unsigned long long)(uintptr_t)gptr;
  asm volatile("global_load_async_to_lds_b128 %0, %1, off"
               :: "v"(lds_addr), "v"(ga)
               : "memory");
}

__device__ __forceinline__ void wait_asynccnt0() {
  asm volatile("s_wait_asynccnt 0x0" ::: "memory");
}

// Core per-element math, shared by tiled + tail kernels.
__device__ __forceinline__ void belief_math(
    float px, float py, float ang,
    float a0, float a1,
    float t0, float t1, float t2,
    const float p[9], float xr[3], float Pr[9]) {
  // range_angle(ang - theta1*a1*DT)
  float angr = ang - t1 * a1 * DT_;
  float ang_ = angr - TWO_PI_ * floorf((angr + PI_) * INV_TWO_PI_);
  float c = __cosf(ang_);
  float s = __sinf(ang_);
  float v = t0 * a0 * DT_;
  float px_ = fminf(fmaxf(fmaf(v, c, px), -1.0f), 1.0f);
  float py_ = fminf(fmaxf(fmaf(v, s, py), -1.0f), 1.0f);

  // A = I + w * e3^T, w = (-v*s, v*c, 0)
  float wx = -v * s;
  float wy = v * c;

  const float p00 = p[0], p01 = p[1], p02 = p[2];
  const float p10 = p[3], p11 = p[4], p12 = p[5];
  const float p20 = p[6], p21 = p[7], p22 = p[8];

  // P' = P + w r^T + c w^T + P22 w w^T  (r = row2, c = col2); then symmetrize.
  float q01 = __expf(2.0f * t2);
  float d00 = p00 + wx * (p20 + p02) + p22 * wx * wx + q01 + 1e-6f;
  float d11 = p11 + wy * (p21 + p12) + p22 * wy * wy + q01 + 1e-6f;
  float d22 = p22 + EXP_M8_ + 1e-6f;
  float s01 = 0.5f * (p01 + p10) + 0.5f * wx * (p21 + p12)
            + 0.5f * wy * (p02 + p20) + p22 * wx * wy;
  float s02 = 0.5f * (p02 + p20) + wx * p22;
  float s12 = 0.5f * (p12 + p21) + wy * p22;

  xr[0] = px_; xr[1] = py_; xr[2] = ang_;
  Pr[0] = d00; Pr[1] = s01; Pr[2] = s02;
  Pr[3] = s01; Pr[4] = d11; Pr[5] = s12;
  Pr[6] = s02; Pr[7] = s12; Pr[8] = d22;
}

// Tiled kernel: one 256-element tile per iteration. Inputs staged to LDS via
// CDNA5 async b128 copies (ASYNCcnt); outputs staged through LDS and written
// with coalesced NT b128 stores (243MB single-use stream > 192MB L2).
__launch_bounds__(BLK)
__global__ void belief_step_tiled(const float* __restrict__ xg,
                                  const float* __restrict__ Pg,
                                  const float* __restrict__ ag,
                                  const float* __restrict__ tg,
                                  float* __restrict__ xo,
                                  float* __restrict__ Po,
                                  int ntiles) {
  __shared__ v4f sP4[(BLK * 9) / 4];  // 576 x 16B
  __shared__ v4f sx4[(BLK * 3) / 4];  // 192
  __shared__ v4f sa4[(BLK * 2) / 4];  // 128
  __shared__ v4f st4[(BLK * 3) / 4];  // 192
  float* sP = (float*)sP4;
  float* sx = (float*)sx4;
  float* sa = (float*)sa4;
  float* st = (float*)st4;

  const int tid = threadIdx.x;

  for (int tile = blockIdx.x; tile < ntiles; tile += gridDim.x) {
    const v4f* P4 = (const v4f*)(Pg + (size_t)tile * (BLK * 9));
    const v4f* x4 = (const v4f*)(xg + (size_t)tile * (BLK * 3));
    const v4f* a4 = (const v4f*)(ag + (size_t)tile * (BLK * 2));
    const v4f* t4 = (const v4f*)(tg + (size_t)tile * (BLK * 3));

    // Prefetch next assigned tile's P region (largest stream).
    int nxt = tile + gridDim.x;
    if (nxt < ntiles) {
      const float* nP = Pg + (size_t)nxt * (BLK * 9);
      __builtin_prefetch(nP + (size_t)tid * 9, 0, 0);  // -> global_prefetch_b8
    }

    // Async coalesced b128 global->LDS staging. 576 = 2*256 + 64.
    async_load_b128_to_lds(P4 + tid,       &sP4[tid]);
    async_load_b128_to_lds(P4 + tid + 256, &sP4[tid + 256]);
    if (tid < 64)  async_load_b128_to_lds(P4 + tid + 512, &sP4[tid + 512]);
    if (tid < 192) async_load_b128_to_lds(x4 + tid, &sx4[tid]);
    if (tid < 128) async_load_b128_to_lds(a4 + tid, &sa4[tid]);
    if (tid < 192) async_load_b128_to_lds(t4 + tid, &st4[tid]);

    // Our async copies must land in LDS before we signal the barrier, so the
    // barrier-complete implies every wave's tile data is visible.
    wait_asynccnt0();
    __syncthreads();

    // Per-thread strided LDS reads (strides 9/3/2 dwords: conflict-free).
    float p[9];
#pragma unroll
    for (int k = 0; k < 9; ++k) p[k] = sP[tid * 9 + k];
    const float px = sx[tid * 3 + 0], py = sx[tid * 3 + 1], ang = sx[tid * 3 + 2];
    const float a0 = sa[tid * 2 + 0], a1 = sa[tid * 2 + 1];
    const float t0 = st[tid * 3 + 0], t1 = st[tid * 3 + 1], t2 = st[tid * 3 + 2];

    float xr[3], Pr[9];
    belief_math(px, py, ang, a0, a1, t0, t1, t2, p, xr, Pr);

    // Each thread writes only its own LDS slots (no cross-thread hazard).
#pragma unroll
    for (int k = 0; k < 9; ++k) sP[tid * 9 + k] = Pr[k];
    sx[tid * 3 + 0] = xr[0]; sx[tid * 3 + 1] = xr[1]; sx[tid * 3 + 2] = xr[2];
    __syncthreads();

    // Cooperative coalesced b128 NT stores from LDS.
    v4f* xo4 = (v4f*)(xo + (size_t)tile * (BLK * 3));
    v4f* Po4 = (v4f*)(Po + (size_t)tile * (BLK * 9));
    __builtin_nontemporal_store(sP4[tid],       Po4 + tid);
    __builtin_nontemporal_store(sP4[tid + 256], Po4 + tid + 256);
    if (tid < 64)  __builtin_nontemporal_store(sP4[tid + 512], Po4 + tid + 512);
    if (tid < 192) __builtin_nontemporal_store(sx4[tid], xo4 + tid);
    __syncthreads();  // all LDS reads done before next iter's async writes
  }
}

// Scalar tail kernel (only used if B % 256 != 0; not hit for B = 2^21).
__global__ void belief_step_tail(const float* __restrict__ xg,
                                 const float* __restrict__ Pg,
                                 const float* __restrict__ ag,
                                 const float* __restrict__ tg,
                                 float* __restrict__ xo,
                                 float* __restrict__ Po,
                                 int base, int B) {
  int i = base + blockIdx.x * blockDim.x + threadIdx.x;
  if (i >= B) return;
  float p[9], xr[3], Pr[9];
#pragma unroll
  for (int k = 0; k < 9; ++k) p[k] = Pg[(size_t)i * 9 + k];
  belief_math(xg[i * 3 + 0], xg[i * 3 + 1], xg[i * 3 + 2],
              ag[i * 2 + 0], ag[i * 2 + 1],
              tg[i * 3 + 0], tg[i * 3 + 1], tg[i * 3 + 2], p, xr, Pr);
#pragma unroll
  for (int k = 0; k < 3; ++k) xo[(size_t)i * 3 + k] = xr[k];
#pragma unroll
  for (int k = 0; k < 9; ++k) Po[(size_t)i * 9 + k] = Pr[k];
}

extern "C" void kernel_launch(void* const* d_in, const int* in_sizes, int n_in,
                              void* d_out, int out_size, void* d_ws, size_t ws_size,
                              hipStream_t stream) {
  const float* x  = (const float*)d_in[0];   // (B,3)
  const float* P  = (const float*)d_in[1];   // (B,3,3)
  const float* a  = (const float*)d_in[2];   // (B,2)
  const float* th = (const float*)d_in[3];   // (B,3)

  const int B = in_sizes[0] / 3;
  float* xo = (float*)d_out;                 // x_ : B*3 floats
  float* Po = xo + (size_t)B * 3;            // P_ : B*9 floats

  const int ntiles = B / BLK;
  if (ntiles > 0) {
    int blocks = ntiles < 4096 ? ntiles : 4096;
    belief_step_tiled<<<blocks, BLK, 0, stream>>>(x, P, a, th, xo, Po, ntiles);
  }
  const int done = ntiles * BLK;
  if (done < B) {
    int rem = B - done;
    belief_step_tail<<<(rem + 255) / 256, 256, 0, stream>>>(x, P, a, th, xo, Po,
                                                            done, B);
  }
}